// Inference_and_Generation_75582834475507
// MI455X (gfx1250) — compile-verified
//
#include <hip/hip_runtime.h>

#define NBOX 4096   // 64*64 boxes per batch
#define NT   1024   // threads per block (32 wave32 waves)
#define PER  4      // boxes per thread

typedef unsigned int u32x4 __attribute__((ext_vector_type(4)));
typedef unsigned int u32x8 __attribute__((ext_vector_type(8)));

__device__ __forceinline__ float sigm(float x) {
    return 1.0f / (1.0f + __expf(-x));
}

// CDNA5 Tensor Data Mover: DMA a contiguous (dim1 x dim0) f32 tile from global
// memory into LDS with a single instruction. D# built per ISA cdna5 §8.3/8.4.
// 2-group form => VADDR2/VADDR3 = NULL (<=2D tensor). Tracked by TENSORcnt.
__device__ __forceinline__ void tdm_load_2d(unsigned lds_addr,
                                            unsigned long long gaddr,
                                            unsigned dim0, unsigned dim1) {
    u32x4 g0;
    g0.x = 1u;                                              // count=1 (valid, user mode)
    g0.y = lds_addr;                                        // D#.lds_addr [63:32]
    g0.z = (unsigned)(gaddr & 0xFFFFFFFFull);               // global_addr[31:0]
    g0.w = (unsigned)((gaddr >> 32) & 0x01FFFFFFull)        // global_addr[56:32]
         | 0x80000000u;                                     // type=2 ("image")
    u32x8 g1;
    g1.s0 = 0x00020000u;                                    // wg_mask=0, data_size=4B
    g1.s1 = (dim0 & 0xFFFFu) << 16;                         // tensor_dim0[15:0]
    g1.s2 = (dim0 >> 16) | ((dim1 & 0xFFFFu) << 16);        // dim0[31:16] | dim1[15:0]
    g1.s3 = (dim1 >> 16) | ((dim0 & 0xFFFFu) << 16);        // dim1[31:16] | tile_dim0
    g1.s4 = (dim1 & 0xFFFFu);                               // tile_dim1 | tile_dim2=0
    g1.s5 = dim0;                                           // tensor_dim0_stride[31:0]
    g1.s6 = 0u;                                             // stride0[47:32] | stride1 lo
    g1.s7 = 0u;                                             // stride1 hi (unused, 2D)
    asm volatile("tensor_load_to_lds %0, %1" :: "s"(g0), "s"(g1) : "memory");
}

// Block-wide argmax with lowest-index tie-break (matches jnp.argmax / stable top_k).
// Wave32 shuffle tree, then LDS tree across the 32 waves.
__device__ __forceinline__ int block_argmax(float v, int idx,
                                            float* s_rval, int* s_ridx, int* s_res,
                                            int t) {
    #pragma unroll
    for (int off = 16; off > 0; off >>= 1) {
        float ov = __shfl_down(v, off, 32);
        int   oi = __shfl_down(idx, off, 32);
        if (ov > v || (ov == v && oi < idx)) { v = ov; idx = oi; }
    }
    if ((t & 31) == 0) { s_rval[t >> 5] = v; s_ridx[t >> 5] = idx; }
    __syncthreads();                       // barrier A
    if (t < 32) {
        float v2 = s_rval[t];
        int   i2 = s_ridx[t];
        #pragma unroll
        for (int off = 16; off > 0; off >>= 1) {
            float ov = __shfl_down(v2, off, 32);
            int   oi = __shfl_down(i2, off, 32);
            if (ov > v2 || (ov == v2 && oi < i2)) { v2 = ov; i2 = oi; }
        }
        if (t == 0) *s_res = i2;
    }
    __syncthreads();                       // barrier B
    return *s_res;
}

__global__ void __launch_bounds__(NT)
nms_topk_fused(const float* __restrict__ tmap, const float* __restrict__ logit,
               const int* __restrict__ nmaxp, const int* __restrict__ topkp,
               float* __restrict__ out)
{
    extern __shared__ float stage[];       // 5 * NBOX floats (tx,ty,tw,th,logit)
    __shared__ float s_rval[32];
    __shared__ int   s_ridx[32];
    __shared__ int   s_res;
    __shared__ float s_bc[4];              // selected box broadcast: bx,by,bw,bh

    const int b = blockIdx.x;              // batch index
    const int t = threadIdx.x;
    const int K = nmaxp[0];                // n_objects_max (device scalar)
    const int topk_only = topkp[0];        // device scalar

    // ---- stage this batch's 80KB slab into LDS via two TDM descriptors ----
    if (t < 32) {                          // one wave issues; TENSORcnt is per-wave
        const unsigned lds_base = (unsigned)(size_t)(&stage[0]);
        tdm_load_2d(lds_base,
                    (unsigned long long)(size_t)(tmap + (size_t)b * 4 * NBOX),
                    NBOX, 4);                          // 4 x 4096 f32 (64KB)
        tdm_load_2d(lds_base + 4u * NBOX * 4u,
                    (unsigned long long)(size_t)(logit + (size_t)b * NBOX),
                    NBOX, 1);                          // 1 x 4096 f32 (16KB)
        asm volatile("s_wait_tensorcnt 0x0" ::: "memory");
    }
    __syncthreads();                       // publish LDS to all 32 waves

    // ---- decode my 4 boxes into registers ----
    float bx[PER], by[PER], bw[PER], bh[PER], prob[PER], chosen[PER], possible[PER];
    #pragma unroll
    for (int m = 0; m < PER; ++m) {
        const int j = t * PER + m;         // list index: ix*64 + iy
        const float tx = sigm(stage[0 * NBOX + j]);
        const float ty = sigm(stage[1 * NBOX + j]);
        const float tw = sigm(stage[2 * NBOX + j]);
        const float th = sigm(stage[3 * NBOX + j]);
        prob[m]        = sigm(stage[4 * NBOX + j]);
        const float ix = (float)(j >> 6);
        const float iy = (float)(j & 63);
        bx[m] = 512.0f * (ix + tx) / 64.0f;
        by[m] = 512.0f * (iy + ty) / 64.0f;
        bw[m] = 10.0f + 30.0f * tw;
        bh[m] = 10.0f + 30.0f * th;
        chosen[m]   = 0.0f;
        possible[m] = 1.0f;
    }

    // ---- greedy NMS: K rounds of argmax + on-the-fly suppression row ----
    if (!topk_only) {
        for (int it = 0; it < K; ++it) {
            float v = prob[0] * possible[0];
            int idx = t * PER;
            #pragma unroll
            for (int m = 1; m < PER; ++m) {
                const float s = prob[m] * possible[m];
                if (s > v) { v = s; idx = t * PER + m; }
            }
            const int sel = block_argmax(v, idx, s_rval, s_ridx, &s_res, t);
            if ((sel >> 2) == t) {                 // owner thread of selected box
                const int m = sel & 3;
                chosen[m] += possible[m];          // onehot * sel (before suppression)
                s_bc[0] = bx[m]; s_bc[1] = by[m];
                s_bc[2] = bw[m]; s_bc[3] = bh[m];
            }
            __syncthreads();                       // barrier C
            const float BX = s_bc[0], BY = s_bc[1], BW = s_bc[2], BH = s_bc[3];
            const float X1 = BX - 0.5f * BW, X3 = BX + 0.5f * BW;
            const float Y1 = BY - 0.5f * BH, Y3 = BY + 0.5f * BH;
            const float AR = BW * BH;
            #pragma unroll
            for (int m = 0; m < PER; ++m) {
                const float x1 = bx[m] - 0.5f * bw[m], x3 = bx[m] + 0.5f * bw[m];
                const float y1 = by[m] - 0.5f * bh[m], y3 = by[m] + 0.5f * bh[m];
                float iw = fminf(X3, x3) - fmaxf(X1, x1); iw = fmaxf(iw, 0.0f);
                float ih = fminf(Y3, y3) - fmaxf(Y1, y1); ih = fmaxf(ih, 0.0f);
                const float inter = iw * ih;
                const float mina  = fmaxf(fminf(AR, bw[m] * bh[m]), 1e-6f);
                if (inter / mina > 0.3f) possible[m] = 0.0f;   // row includes diagonal
            }
        }
    }

    // ---- top-k via K rounds of stable argmax; owner writes outputs ----
    float score[PER];
    #pragma unroll
    for (int m = 0; m < PER; ++m)
        score[m] = topk_only ? prob[m] : prob[m] * chosen[m];

    for (int k = 0; k < K; ++k) {
        float v = score[0];
        int idx = t * PER;
        #pragma unroll
        for (int m = 1; m < PER; ++m) {
            if (score[m] > v) { v = score[m]; idx = t * PER + m; }
        }
        const int sel = block_argmax(v, idx, s_rval, s_ridx, &s_res, t);
        if ((sel >> 2) == t) {
            const int m = sel & 3;
            const int KB = K * 4;                  // B = gridDim.x = 4 batches
            out[0 * KB + k * 4 + b] = score[m];    // prob_few (masked prob)
            out[1 * KB + k * 4 + b] = bx[m];
            out[2 * KB + k * 4 + b] = by[m];
            out[3 * KB + k * 4 + b] = bw[m];
            out[4 * KB + k * 4 + b] = bh[m];
            score[m] = -1.0f;                      // exclude; all live values >= 0
        }
    }
}

extern "C" void kernel_launch(void* const* d_in, const int* in_sizes, int n_in,
                              void* d_out, int out_size, void* d_ws, size_t ws_size,
                              hipStream_t stream) {
    const float* tmap  = (const float*)d_in[0];
    const float* logit = (const float*)d_in[1];
    const int*   nmax  = (const int*)d_in[2];
    const int*   topk  = (const int*)d_in[3];
    float* out = (float*)d_out;

    const int B = in_sizes[0] / (4 * NBOX);        // 4 batches
    const size_t shmem = (size_t)5 * NBOX * sizeof(float);   // 80 KB dynamic LDS
    nms_topk_fused<<<dim3(B), dim3(NT), shmem, stream>>>(tmap, logit, nmax, topk, out);
}